// Grouping_37022618092259
// MI455X (gfx1250) — compile-verified
//
#include <hip/hip_runtime.h>
#include <hip/hip_bf16.h>
#include <math.h>

#define B_   16
#define N_   8192
#define CIN_ 64
#define S_   512
#define K_   32
#define DIN_ 134      // 2*CIN+6
#define DPAD 136      // padded to multiple of 4 for wmma f32 16x16x4

typedef __attribute__((ext_vector_type(2))) float v2f;
typedef __attribute__((ext_vector_type(8))) float v8f;

// ---------------------------------------------------------------------------
// Zero the per-batch moment accumulators.
// ---------------------------------------------------------------------------
__global__ void init_sums_kernel(float* __restrict__ sums) {
    if (threadIdx.x < 2 * B_) sums[threadIdx.x] = 0.0f;
}

// ---------------------------------------------------------------------------
// Farthest point sampling: one block per batch, 256 threads, 32 points/thread.
// dist[] lives in registers; block argmax via LDS tree reduction.
// Emits new_idx[b][s] (first pick = 0, matching the scan's output-of-carry)
// and new_xyz.
// ---------------------------------------------------------------------------
__global__ __launch_bounds__(256) void fps_kernel(const float* __restrict__ xyz,
                                                  int* __restrict__ new_idx,
                                                  float* __restrict__ new_xyz) {
    const int b = blockIdx.x;
    const int t = threadIdx.x;
    const float* xb = xyz + (size_t)b * N_ * 3;

    float dist[N_ / 256];
#pragma unroll
    for (int j = 0; j < N_ / 256; ++j) dist[j] = 1e10f;

    __shared__ float sp[3];
    __shared__ float rv[256];
    __shared__ int   ri[256];
    __shared__ int   s_last;
    if (t == 0) s_last = 0;
    __syncthreads();

    for (int s = 0; s < S_; ++s) {
        if (t == 0) {
            int last = s_last;
            float px = xb[last * 3 + 0];
            float py = xb[last * 3 + 1];
            float pz = xb[last * 3 + 2];
            sp[0] = px; sp[1] = py; sp[2] = pz;
            new_idx[b * S_ + s] = last;
            new_xyz[((size_t)b * S_ + s) * 3 + 0] = px;
            new_xyz[((size_t)b * S_ + s) * 3 + 1] = py;
            new_xyz[((size_t)b * S_ + s) * 3 + 2] = pz;
        }
        __syncthreads();
        const float px = sp[0], py = sp[1], pz = sp[2];

        float bestv = -1.0f;
        int   besti = 0x7fffffff;
#pragma unroll
        for (int j = 0; j < N_ / 256; ++j) {
            const int i = t + j * 256;
            const float dx = xb[i * 3 + 0] - px;
            const float dy = xb[i * 3 + 1] - py;
            const float dz = xb[i * 3 + 2] - pz;
            float d = dx * dx + dy * dy + dz * dz;
            float nd = fminf(dist[j], d);
            dist[j] = nd;
            if (nd > bestv) { bestv = nd; besti = i; }   // ascending i => lowest idx wins ties
        }
        rv[t] = bestv; ri[t] = besti;
        __syncthreads();
        for (int off = 128; off > 0; off >>= 1) {
            if (t < off) {
                const float v2 = rv[t + off];
                const int   i2 = ri[t + off];
                if (v2 > rv[t] || (v2 == rv[t] && i2 < ri[t])) { rv[t] = v2; ri[t] = i2; }
            }
            __syncthreads();
        }
        if (t == 0) s_last = ri[0];
        __syncthreads();
    }
}

// ---------------------------------------------------------------------------
// KNN (one thread per centroid, register-resident sorted top-32), gather,
// grouped_xyz / grouped_xyz_local / mask outputs, and per-batch moment
// accumulation for the ddof=1 std over the 70 pre-normalization channels.
// ---------------------------------------------------------------------------
__global__ __launch_bounds__(256) void knn_kernel(const float* __restrict__ xyz,
                                                  const float* __restrict__ f,
                                                  const int* __restrict__ new_idx,
                                                  int* __restrict__ grouped_idx,
                                                  float* __restrict__ out_gxyz,
                                                  float* __restrict__ out_glocal,
                                                  float* __restrict__ out_mask,
                                                  float* __restrict__ sums) {
    const int gid = blockIdx.x * blockDim.x + threadIdx.x;
    if (gid >= B_ * S_) return;
    const int b = gid / S_;
    const float* xb = xyz + (size_t)b * N_ * 3;
    const float* fb = f + (size_t)b * N_ * CIN_;

    const int ci = new_idx[gid];
    const float cx = xb[ci * 3 + 0];
    const float cy = xb[ci * 3 + 1];
    const float cz = xb[ci * 3 + 2];

    float bd[K_];
    int   bi[K_];
#pragma unroll
    for (int k = 0; k < K_; ++k) { bd[k] = 3.4e38f; bi[k] = 0; }

    for (int i = 0; i < N_; ++i) {
        const float dx = xb[i * 3 + 0] - cx;
        const float dy = xb[i * 3 + 1] - cy;
        const float dz = xb[i * 3 + 2] - cz;
        const float d = dx * dx + dy * dy + dz * dz;
        if (d < bd[K_ - 1]) {                 // strict: equal dists keep earlier index
            bool ins = false;
#pragma unroll
            for (int j = K_ - 1; j > 0; --j) {
                const float pv = bd[j - 1];
                const int   pi = bi[j - 1];
                const bool shift = (!ins) && (pv > d);
                const bool place = (!ins) && !(pv > d);
                if (shift) { bd[j] = pv; bi[j] = pi; }
                if (place) { bd[j] = d; bi[j] = i; ins = true; }
            }
            if (!ins) { bd[0] = d; bi[0] = i; }
        }
    }

    const size_t base = (size_t)gid * K_;
#pragma unroll
    for (int k = 0; k < K_; ++k) grouped_idx[base + k] = bi[k];

    float lsum = 0.0f, lsq = 0.0f;
    for (int k = 0; k < K_; ++k) {
        const int gi = grouped_idx[base + k];
        const float gx = xb[gi * 3 + 0];
        const float gy = xb[gi * 3 + 1];
        const float gz = xb[gi * 3 + 2];
        const float lx = gx - cx, ly = gy - cy, lz = gz - cz;
        out_gxyz[(base + k) * 3 + 0] = gx;
        out_gxyz[(base + k) * 3 + 1] = gy;
        out_gxyz[(base + k) * 3 + 2] = gz;
        out_glocal[(base + k) * 3 + 0] = lx;
        out_glocal[(base + k) * 3 + 1] = ly;
        out_glocal[(base + k) * 3 + 2] = lz;
        out_mask[base + k] = 0.0f;
        lsum += gx + gy + gz + lx + ly + lz;
        lsq  += gx * gx + gy * gy + gz * gz + lx * lx + ly * ly + lz * lz;
        for (int c = 0; c < CIN_; ++c) {
            const float v = fb[gi * CIN_ + c] - fb[ci * CIN_ + c];
            lsum += v;
            lsq  += v * v;
        }
    }
    atomicAdd(&sums[b], lsum);
    atomicAdd(&sums[B_ + b], lsq);
}

// ---------------------------------------------------------------------------
// Per-batch scale = 1 / (std(ddof=1) + 1e-5)
// ---------------------------------------------------------------------------
__global__ void stats_kernel(const float* __restrict__ sums, float* __restrict__ scale) {
    const int b = threadIdx.x;
    if (b < B_) {
        const double n = (double)S_ * K_ * (CIN_ + 6);
        const double mean = (double)sums[b] / n;
        double var = ((double)sums[B_ + b] - n * mean * mean) / (n - 1.0);
        if (var < 0.0) var = 0.0;
        const float sd = (float)sqrt(var);
        scale[b] = 1.0f / (sd + 1e-5f);
    }
}

// ---------------------------------------------------------------------------
// Final mix: grouped_f[B*S*K, 134] @ mix_w^T -> [B*S*K, 64] via
// V_WMMA_F32_16X16X4_F32 (full fp32, matches reference precision).
// 128-thread WG: stage a 64x136 activation tile into LDS, then 4 waves each
// compute a 16-row x 64-col output tile as 4 col-tiles x 34 k-steps.
// ---------------------------------------------------------------------------
__global__ __launch_bounds__(128) void mix_kernel(const float* __restrict__ xyz,
                                                  const float* __restrict__ f,
                                                  const int* __restrict__ new_idx,
                                                  const int* __restrict__ grouped_idx,
                                                  const float* __restrict__ aw,
                                                  const float* __restrict__ ab,
                                                  const float* __restrict__ mix_w,
                                                  const float* __restrict__ mix_b,
                                                  const float* __restrict__ scale,
                                                  float* __restrict__ out_gf) {
    __shared__ float A[4][16][DPAD];

    const int tid = threadIdx.x;
    const int row0 = blockIdx.x * 64;           // 64 rows per WG; S*K divisible by 64

    // Phase 1: cooperative tile fill (68 elements per thread, no divergence).
    for (int idx = tid; idx < 64 * DPAD; idx += 128) {
        const int r = idx / DPAD;
        const int d = idx % DPAD;
        const int gr = row0 + r;
        const int b = gr / (S_ * K_);
        const int rem = gr % (S_ * K_);
        const int s = rem / K_;
        const int ci = new_idx[b * S_ + s];
        const int gi = grouped_idx[gr];
        const float* xb = xyz + (size_t)b * N_ * 3;
        const float* fb = f + (size_t)b * N_ * CIN_;
        const float sc = scale[b];
        float val;
        if (d < 3) {
            val = aw[d] * (xb[gi * 3 + d] * sc) + ab[d];
        } else if (d < 6) {
            val = aw[d] * ((xb[gi * 3 + d - 3] - xb[ci * 3 + d - 3]) * sc) + ab[d];
        } else if (d < 70) {
            const int c = d - 6;
            val = aw[d] * ((fb[gi * CIN_ + c] - fb[ci * CIN_ + c]) * sc) + ab[d];
        } else if (d < DIN_) {
            val = fb[ci * CIN_ + (d - 70)];     // sampled_rep, un-normalized
        } else {
            val = 0.0f;                          // k-padding
        }
        A[r >> 4][r & 15][d] = val;
    }
    __syncthreads();

    // Phase 2: WMMA. EXEC is all-ones (no guards, exact grid).
    const int w    = tid >> 5;      // wave id 0..3
    const int lane = tid & 31;
    const int half = lane >> 4;     // 0: K={0,1}, 1: K={2,3}
    const int lr   = lane & 15;     // row (A) / col (B,D) within tile
    const int trow0 = row0 + w * 16;

    for (int ct = 0; ct < 4; ++ct) {
        v8f acc = {0.f, 0.f, 0.f, 0.f, 0.f, 0.f, 0.f, 0.f};
        const int col = ct * 16 + lr;
#pragma unroll
        for (int step = 0; step < DPAD / 4; ++step) {
            const int k0 = step * 4 + half * 2;
            v2f av;
            av.x = A[w][lr][k0 + 0];
            av.y = A[w][lr][k0 + 1];
            v2f bv;                              // B[k][n] = mix_w[n][k]
            bv.x = (k0 + 0 < DIN_) ? mix_w[col * DIN_ + k0 + 0] : 0.0f;
            bv.y = (k0 + 1 < DIN_) ? mix_w[col * DIN_ + k0 + 1] : 0.0f;
            acc = __builtin_amdgcn_wmma_f32_16x16x4_f32(
                /*neg_a=*/false, av, /*neg_b=*/false, bv,
                /*c_mod=*/(short)0, acc, /*reuse_a=*/false, /*reuse_b=*/false);
        }
        const float bias = mix_b[col];
#pragma unroll
        for (int j = 0; j < 8; ++j) {
            const int m = j + half * 8;          // D: VGPR j -> M = j + 8*(lane>=16)
            out_gf[(size_t)(trow0 + m) * CIN_ + col] = acc[j] + bias;
        }
    }
}

// ---------------------------------------------------------------------------
// Launch
// ---------------------------------------------------------------------------
extern "C" void kernel_launch(void* const* d_in, const int* in_sizes, int n_in,
                              void* d_out, int out_size, void* d_ws, size_t ws_size,
                              hipStream_t stream) {
    const float* xyz   = (const float*)d_in[0];   // [16,8192,3]
    const float* f     = (const float*)d_in[1];   // [16,8192,64]
    const float* aw    = (const float*)d_in[2];   // [70]
    const float* ab    = (const float*)d_in[3];   // [70]
    const float* mix_w = (const float*)d_in[4];   // [64,134]
    const float* mix_b = (const float*)d_in[5];   // [64]

    float* out = (float*)d_out;
    float* o_newxyz = out;                                        // 16*512*3      = 24576
    float* o_gxyz   = o_newxyz + (size_t)B_ * S_ * 3;             // 16*512*32*3   = 786432
    float* o_glocal = o_gxyz   + (size_t)B_ * S_ * K_ * 3;        // 786432
    float* o_gf     = o_glocal + (size_t)B_ * S_ * K_ * 3;        // 16*512*32*64  = 16777216
    float* o_mask   = o_gf     + (size_t)B_ * S_ * K_ * CIN_;     // 262144

    int*   ws_newidx = (int*)d_ws;                                // B*S ints
    int*   ws_gidx   = ws_newidx + B_ * S_;                       // B*S*K ints
    float* ws_sums   = (float*)(ws_gidx + B_ * S_ * K_);          // 2*B floats
    float* ws_scale  = ws_sums + 2 * B_;                          // B floats

    init_sums_kernel<<<1, 64, 0, stream>>>(ws_sums);
    fps_kernel<<<B_, 256, 0, stream>>>(xyz, ws_newidx, o_newxyz);
    knn_kernel<<<(B_ * S_ + 255) / 256, 256, 0, stream>>>(
        xyz, f, ws_newidx, ws_gidx, o_gxyz, o_glocal, o_mask, ws_sums);
    stats_kernel<<<1, 32, 0, stream>>>(ws_sums, ws_scale);
    mix_kernel<<<(B_ * S_ * K_) / 64, 128, 0, stream>>>(
        xyz, f, ws_newidx, ws_gidx, aw, ab, mix_w, mix_b, ws_scale, o_gf);
}